// FocalLoss_18897856103311
// MI455X (gfx1250) — compile-verified
//
#include <hip/hip_runtime.h>
#include <hip/hip_bf16.h>
#include <stdint.h>

// ---------------------------------------------------------------------------
// RetinaNet focal loss, fused single pass over the 205MB classification tensor.
// MI455X (gfx1250): wave32, WMMA reduction, TDM annotation staging.
// ---------------------------------------------------------------------------

typedef __attribute__((ext_vector_type(2))) float        v2f;
typedef __attribute__((ext_vector_type(8))) float        v8f;
typedef __attribute__((ext_vector_type(4))) unsigned int v4u;
typedef __attribute__((ext_vector_type(4))) int          v4i;
typedef __attribute__((ext_vector_type(8))) int          v8i;

#define FL_ALPHA 0.25f
#define FL_EPS   1e-4f
#define MAXM     64      // LDS annotation capacity (actual M = 32)
#define NTHR     256     // 8 waves / block

// --- wave32 full reduction using V_WMMA_F32_16X16X4_F32 ---------------------
// A (16x4) = all ones (every lane supplies {1,1}); B (4x16): VGPR0 = lane
// value (rows K=0,1), VGPR1 = 0 (rows K=2,3).  D[m][n] = x[n] + x[n+16],
// so d[0] holds the pairwise half-wave sums; 4 xor-shuffles finish the sum.
__device__ __forceinline__ float wave_sum32(float x) {
    v2f a; a.x = 1.0f; a.y = 1.0f;
    v2f b; b.x = x;    b.y = 0.0f;
    v8f c = {};
    c = __builtin_amdgcn_wmma_f32_16x16x4_f32(
            /*neg_a=*/false, a, /*neg_b=*/false, b,
            /*c_mod=*/(short)0, c, /*reuse_a=*/false, /*reuse_b=*/false);
    float s = c[0];               // = x[n] + x[n+16], n = lane & 15
    s += __shfl_xor(s, 8);
    s += __shfl_xor(s, 4);
    s += __shfl_xor(s, 2);
    s += __shfl_xor(s, 1);
    return s;                     // full 32-lane sum in every lane
}

__global__ void fl_init(float* ws, int n) {
    int i = blockIdx.x * blockDim.x + threadIdx.x;
    if (i < n) ws[i] = 0.0f;
}

__global__ __launch_bounds__(NTHR)
void fl_main(const float* __restrict__ cls,   // (B, A, K)
             const float* __restrict__ reg,   // (B, A, 4)
             const float* __restrict__ anch,  // (1, A, 4)
             const float* __restrict__ ann,   // (B, M, 5)
             float* __restrict__ ws,          // [0,B) cls_sum  [B,2B) reg_sum  [2B,3B) npos
             int Acnt, int K, int M)
{
    __shared__ float s_ann[MAXM * 5];
    __shared__ float s_red[NTHR / 32][3];

    const int b    = blockIdx.y;
    const int tid  = threadIdx.x;
    const int lane = tid & 31;
    const int wid  = tid >> 5;

    // ---- stage annotations[b] (M*5 floats) into LDS via the Tensor DMA ----
    if (wid == 0) {
        const unsigned n0 = (unsigned)(M * 5);                 // elements (f32)
        unsigned long long ga =
            (unsigned long long)(uintptr_t)(ann + (size_t)b * n0);
        unsigned ldsa = (unsigned)(uintptr_t)(void*)&s_ann[0]; // low 32b = LDS offset

        v4u g0;
        g0[0] = 1u;                                            // count=1, user mode
        g0[1] = ldsa;                                          // lds_addr
        g0[2] = (unsigned)(ga & 0xffffffffu);                  // global_addr[31:0]
        g0[3] = (unsigned)((ga >> 32) & 0x1ffffffu)            // global_addr[56:32]
              | 0x80000000u;                                   // type = 2 ("image")

        v8i g1;
        g1[0] = (int)(2u << 16);                 // wg_mask=0, data_size=2 (4B)
        g1[1] = (int)(n0 << 16);                 // tensor_dim0[15:0]
        g1[2] = (int)((n0 >> 16) | (1u << 16)); // tensor_dim0[31:16] | tensor_dim1=1
        g1[3] = (int)(n0 << 16);                 // tile_dim0 = n0
        g1[4] = 1;                               // tile_dim1 = 1, tile_dim2 = 0
        g1[5] = (int)n0;                         // tensor_dim0_stride[31:0]
        g1[6] = 0;                               // stride0 hi | stride1 lo
        g1[7] = 0;

        v4i gz;  gz[0] = 0; gz[1] = 0; gz[2] = 0; gz[3] = 0;
        v8i gz8; gz8[0]=0; gz8[1]=0; gz8[2]=0; gz8[3]=0;
                 gz8[4]=0; gz8[5]=0; gz8[6]=0; gz8[7]=0;

        // amdgpu-toolchain (clang-23) 6-arg form:
        // (v4u g0, v8i g1, v4i g2, v4i g3, v8i extra, i32 cpol)
        __builtin_amdgcn_tensor_load_to_lds(g0, g1, gz, gz, gz8, /*cpol=*/0);
        __builtin_amdgcn_s_wait_tensorcnt(0);
    }
    __syncthreads();

    float cls_acc = 0.0f, reg_acc = 0.0f, np_acc = 0.0f;

    const int a = blockIdx.x * NTHR + tid;
    if (a < Acnt) {
        const float4 av = ((const float4*)anch)[a];
        const float ax1 = av.x, ay1 = av.y, ax2 = av.z, ay2 = av.w;
        const float aw = ax2 - ax1, ah = ay2 - ay1;
        const float acx = ax1 + 0.5f * aw, acy = ay1 + 0.5f * ah;
        const float areaA = aw * ah;

        // ---- IoU max / argmax over annotations (broadcast from LDS) ----
        float best = -INFINITY; int bestm = 0;
        for (int m = 0; m < M; ++m) {
            const float bx1 = s_ann[m*5+0], by1 = s_ann[m*5+1];
            const float bx2 = s_ann[m*5+2], by2 = s_ann[m*5+3];
            const float lb  = s_ann[m*5+4];
            float iw = fminf(ax2, bx2) - fmaxf(ax1, bx1); iw = fmaxf(iw, 0.0f);
            float ih = fminf(ay2, by2) - fmaxf(ay1, by1); ih = fmaxf(ih, 0.0f);
            const float inter = iw * ih;
            const float ua = fmaxf(areaA + (bx2-bx1)*(by2-by1) - inter, 1e-8f);
            float iou = inter / ua;
            iou = (lb != -1.0f) ? iou : -1.0f;
            if (iou > best) { best = iou; bestm = m; }   // first max wins (argmax)
        }
        const bool pos = best >= 0.5f;
        const bool neg = best <  0.4f;

        // ---- smooth-L1 regression loss (positives only) ----
        if (pos) {
            const float gx1 = s_ann[bestm*5+0], gy1 = s_ann[bestm*5+1];
            const float gx2 = s_ann[bestm*5+2], gy2 = s_ann[bestm*5+3];
            const float gw = fmaxf(gx2 - gx1, 1.0f);
            const float gh = fmaxf(gy2 - gy1, 1.0f);
            const float gcx = gx1 + 0.5f * gw, gcy = gy1 + 0.5f * gh;
            const float t0 = (gcx - acx) / aw * 10.0f;   // / BBOX_STD {0.1,0.1,0.2,0.2}
            const float t1 = (gcy - acy) / ah * 10.0f;
            const float t2 = __logf(gw / aw) * 5.0f;
            const float t3 = __logf(gh / ah) * 5.0f;
            const float4 rv = ((const float4*)reg)[(size_t)b * Acnt + a];
            const float d0 = fabsf(t0 - rv.x), d1 = fabsf(t1 - rv.y);
            const float d2 = fabsf(t2 - rv.z), d3 = fabsf(t3 - rv.w);
            #define SL1(d) ((d) <= (1.0f/9.0f) ? 4.5f*(d)*(d) : (d) - (1.0f/18.0f))
            reg_acc = SL1(d0) + SL1(d1) + SL1(d2) + SL1(d3);
            #undef SL1
            np_acc = 1.0f;
        }

        // ---- focal classification loss over K classes (pos|neg anchors) ----
        if (pos || neg) {
            const int labi = pos ? (int)s_ann[bestm*5+4] : -1;
            const float* row = cls + ((size_t)b * Acnt + a) * (size_t)K;
            float s = 0.0f;
            int k = 0;
            for (; k + 3 < K; k += 4) {
                const float4 p4 = *(const float4*)(row + k);
                #define FTERM(pr, kk)                                          \
                    {                                                          \
                        float p  = fminf(fmaxf((pr), FL_EPS), 1.0f - FL_EPS);  \
                        bool  hit = ((kk) == labi);                            \
                        float pm = hit ? 1.0f - p : p;                         \
                        float af = hit ? FL_ALPHA : 1.0f - FL_ALPHA;           \
                        s += af * pm * pm * (-__logf(1.0f - pm));              \
                    }
                FTERM(p4.x, k + 0)
                FTERM(p4.y, k + 1)
                FTERM(p4.z, k + 2)
                FTERM(p4.w, k + 3)
            }
            for (; k < K; ++k) { FTERM(row[k], k) }
            #undef FTERM
            cls_acc = s;
        }
    }

    // ---- wave reduction via WMMA (EXEC all-ones here: fully reconverged) ----
    const float wc = wave_sum32(cls_acc);
    const float wr = wave_sum32(reg_acc);
    const float wn = wave_sum32(np_acc);
    if (lane == 0) { s_red[wid][0] = wc; s_red[wid][1] = wr; s_red[wid][2] = wn; }
    __syncthreads();

    if (tid == 0) {
        float c = 0.0f, r = 0.0f, n = 0.0f;
        for (int w = 0; w < NTHR / 32; ++w) {
            c += s_red[w][0]; r += s_red[w][1]; n += s_red[w][2];
        }
        const int B = gridDim.y;
        atomicAdd(&ws[b],          c);
        atomicAdd(&ws[B + b],      r);
        atomicAdd(&ws[2 * B + b],  n);
    }
}

__global__ void fl_final(const float* __restrict__ ws,
                         const float* __restrict__ ann,
                         float* __restrict__ out, int B, int M)
{
    const int t = threadIdx.x;   // 32 threads
    float clsv = 0.0f, regv = 0.0f;
    if (t < B) {
        bool has = false;
        for (int m = 0; m < M; ++m)
            has = has || (ann[(size_t)t * M * 5 + m * 5 + 4] != -1.0f);
        const float np = ws[2 * B + t];
        const float c  = ws[t]     / fmaxf(np,        1.0f);
        const float r  = ws[B + t] / fmaxf(np * 4.0f, 1.0f);
        clsv = has ? c : 0.0f;
        regv = has ? r : 0.0f;
    }
    for (int off = 16; off > 0; off >>= 1) {
        clsv += __shfl_down(clsv, off);
        regv += __shfl_down(regv, off);
    }
    if (t == 0) { out[0] = clsv / (float)B; out[1] = regv / (float)B; }
}

extern "C" void kernel_launch(void* const* d_in, const int* in_sizes, int n_in,
                              void* d_out, int out_size, void* d_ws, size_t ws_size,
                              hipStream_t stream) {
    const float* cls  = (const float*)d_in[0];   // (B, A, K)
    const float* reg  = (const float*)d_in[1];   // (B, A, 4)
    const float* anch = (const float*)d_in[2];   // (1, A, 4)
    const float* ann  = (const float*)d_in[3];   // (B, M, 5)

    const int A  = in_sizes[2] / 4;
    const int Bn = in_sizes[1] / (4 * A);
    const int K  = in_sizes[0] / (Bn * A);
    int       M  = in_sizes[3] / (5 * Bn);
    if (M > MAXM) M = MAXM;

    float* ws = (float*)d_ws;

    fl_init<<<1, 64, 0, stream>>>(ws, 3 * Bn);

    dim3 grid((A + NTHR - 1) / NTHR, Bn);
    fl_main<<<grid, NTHR, 0, stream>>>(cls, reg, anch, ann, ws, A, K, M);

    fl_final<<<1, 32, 0, stream>>>(ws, ann, (float*)d_out, Bn, M);
}